// SegmentationBranch_55740085567519
// MI455X (gfx1250) — compile-verified
//
#include <hip/hip_runtime.h>
#include <hip/hip_bf16.h>

typedef __attribute__((ext_vector_type(2))) float v2f;
typedef __attribute__((ext_vector_type(8))) float v8f;
typedef __attribute__((ext_vector_type(4))) unsigned int v4u;
typedef __attribute__((ext_vector_type(8))) int v8i;
typedef __attribute__((ext_vector_type(4))) int v4i;

#define NC   10
#define HIDD 64
#define CINN 512
#define B_   4
#define H1   60
#define W1_  80
#define HW1  (H1 * W1_)      // 4800
#define H2   30
#define W2_  40
#define HW2  (H2 * W2_)      // 1200
#define OH   480
#define OW   640
#define OHW  (OH * OW)       // 307200
#define NCL  (NC + 1)        // 11 head channels

// ---------------------------------------------------------------------------
// Tensor Data Mover: 2-D tile Global -> LDS.  D# built per CDNA5 ISA.
//   group0: count=1 | lds_addr | global_addr[56:0] | type=2
//   group1: data_size | tensor_dim0/1 | tile_dim0/1 | tensor_dim0_stride
//   groups 2/3 (+ extra group on this toolchain): zero (2-D, tile_dim2 = 0)
// OOB reads (row >= tensor_d1) return ZERO -> free zero-padding.
// This toolchain exposes the 6-arg builtin:
//   (v4u g0, v8i g1, v4i g2, v4i g3, v8i g4, i32 cpol)
// ---------------------------------------------------------------------------
__device__ __forceinline__ void tdm_load_2d(const void* gptr, unsigned int lds_addr,
                                            unsigned int tensor_d0, unsigned int tensor_d1,
                                            unsigned long long stride0_elems,
                                            unsigned int tile_d0, unsigned int tile_d1,
                                            unsigned int data_size_code /*2 = 4B*/) {
    unsigned long long ga = (unsigned long long)gptr;
    v4u g0;
    g0[0] = 1u;                                                   // count=1 (user)
    g0[1] = lds_addr;                                             // LDS byte address
    g0[2] = (unsigned int)ga;                                     // global_addr[31:0]
    g0[3] = (unsigned int)((ga >> 32) & 0x01FFFFFFu) | (2u << 30);// [56:32] | type=2
    v8i g1;
    g1[0] = (int)(data_size_code << 16);                          // wg_mask=0 | data_size
    g1[1] = (int)((tensor_d0 & 0xFFFFu) << 16);                   // abar=0 | td0[15:0]
    g1[2] = (int)(((tensor_d1 & 0xFFFFu) << 16) | ((tensor_d0 >> 16) & 0xFFFFu));
    g1[3] = (int)(((tile_d0  & 0xFFFFu) << 16) | ((tensor_d1 >> 16) & 0xFFFFu));
    g1[4] = (int)(tile_d1 & 0xFFFFu);                             // tile_dim2 = 0 (2-D)
    g1[5] = (int)(unsigned int)(stride0_elems & 0xFFFFFFFFull);
    g1[6] = (int)(unsigned int)((stride0_elems >> 32) & 0xFFFFull); // dim1_stride=0
    g1[7] = 0;
    v4i z4 = {0, 0, 0, 0};
    v8i z8 = {0, 0, 0, 0, 0, 0, 0, 0};
    __builtin_amdgcn_tensor_load_to_lds(g0, g1, z4, z4, z8, 0);
}

// ---------------------------------------------------------------------------
// Stage 1/2: 1x1 conv (channel GEMM) + bias + relu, V_WMMA_F32_16X16X4_F32.
// Per wave: M=64 (4 tiles) x N=32 pixels (2 tiles), K=512 in steps of 4.
// Weights staged through LDS by the Tensor Data Mover in 32KB K-chunks.
// ---------------------------------------------------------------------------
#define KC 128   // K-chunk in LDS: 64*128*4B = 32KB

__global__ __launch_bounds__(256)
void conv1x1_relu_wmma(const float* __restrict__ feat,   // [B][CIN][HW]
                       const float* __restrict__ W,      // [HID][CIN]
                       const float* __restrict__ bias,   // [HID]
                       float* __restrict__ out,          // [B][HID][HW]
                       int HW) {
    __shared__ float Ws[HIDD * KC];  // 32 KB

    const int tid  = threadIdx.x;
    const int b    = blockIdx.y;
    const int wave = tid >> 5;
    const int lane = tid & 31;
    const int half = lane >> 4;   // 0: lanes 0-15, 1: lanes 16-31
    const int lr   = lane & 15;

    // pixel base for this wave; clamp (duplicate work) instead of returning so
    // EXEC stays all-1s for WMMA and every wave reaches the barriers.
    int p0 = blockIdx.x * (8 * 32) + wave * 32;
    if (p0 + 32 > HW) p0 = HW - 32;

    const float* X = feat + (size_t)b * CINN * HW;
    const unsigned int lds_base = (unsigned int)(size_t)(void*)Ws;

    v8f acc[8] = {v8f{}, v8f{}, v8f{}, v8f{}, v8f{}, v8f{}, v8f{}, v8f{}};

    for (int kc = 0; kc < CINN; kc += KC) {
        __syncthreads();   // previous chunk fully consumed before TDM overwrites
        if (wave == 0) {
            // DMA W[0:64][kc:kc+KC] -> LDS, row-major width KC
            tdm_load_2d(W + kc, lds_base,
                        /*tensor_d0=*/KC, /*tensor_d1=*/HIDD,
                        /*stride0=*/CINN, /*tile_d0=*/KC, /*tile_d1=*/HIDD, 2u);
            __builtin_amdgcn_s_wait_tensorcnt(0);
        }
        __syncthreads();

        if (kc + KC < CINN) {  // pull next chunk's activations toward L2
            __builtin_prefetch(X + (size_t)(kc + KC) * HW + p0 + lane, 0, 1);
        }

        // induction-variable addressing: xrow advances 4 rows (4*HW floats)/step
        const float* xrow = X + (size_t)(kc + 2 * half) * HW + p0 + lr;
        for (int k = 0; k < KC; k += 4) {
            const int kk = k + 2 * half;
            v2f bf0, bf1;
            bf0[0] = xrow[0];   bf0[1] = xrow[HW];        // cols p0+lr
            bf1[0] = xrow[16];  bf1[1] = xrow[HW + 16];   // cols p0+16+lr
#pragma unroll
            for (int mt = 0; mt < 4; ++mt) {
                const float* wr = &Ws[(mt * 16 + lr) * KC + kk];
                v2f a;
                a[0] = wr[0];
                a[1] = wr[1];
                acc[mt] = __builtin_amdgcn_wmma_f32_16x16x4_f32(
                    false, a, false, bf0, (short)0, acc[mt], false, false);
                acc[mt + 4] = __builtin_amdgcn_wmma_f32_16x16x4_f32(
                    false, a, false, bf1, (short)0, acc[mt + 4], false, false);
            }
            xrow += 4 * (size_t)HW;
        }
    }

    // epilogue: bias + relu.  D layout: VGPR v -> row = mt*16 + v + 8*half
    float* Yo = out + (size_t)b * HIDD * HW;
#pragma unroll
    for (int mt = 0; mt < 4; ++mt) {
#pragma unroll
        for (int v = 0; v < 8; ++v) {
            const int row = mt * 16 + v + 8 * half;
            const float bv = bias[row];
            float* yr = Yo + (size_t)row * HW + p0 + lr;
            yr[0]  = fmaxf(acc[mt][v] + bv, 0.0f);
            yr[16] = fmaxf(acc[mt + 4][v] + bv, 0.0f);
        }
    }
}

// ---------------------------------------------------------------------------
// Stage 3a: inter = hid1 + bilinear_x2(hid2)   (64 ch @ 60x80)
// ---------------------------------------------------------------------------
__global__ __launch_bounds__(256)
void upsample2x_add(const float* __restrict__ hid1,   // [B*64][60][80]
                    const float* __restrict__ hid2,   // [B*64][30][40]
                    float* __restrict__ inter) {      // [B*64][60][80]
    const int idx = blockIdx.x * 256 + threadIdx.x;
    const int total = B_ * HIDD * HW1;
    if (idx >= total) return;
    const int p  = idx % HW1;
    const int bc = idx / HW1;
    const int x = p % W1_, y = p / W1_;

    float sx = (x + 0.5f) * 0.5f - 0.5f;
    float sy = (y + 0.5f) * 0.5f - 0.5f;
    int ix0 = (int)floorf(sx);  float fx = sx - (float)ix0;
    int iy0 = (int)floorf(sy);  float fy = sy - (float)iy0;
    int ix1 = ix0 + 1, iy1 = iy0 + 1;
    ix0 = min(max(ix0, 0), W2_ - 1); ix1 = min(max(ix1, 0), W2_ - 1);
    iy0 = min(max(iy0, 0), H2 - 1);  iy1 = min(max(iy1, 0), H2 - 1);

    const float* S = hid2 + (size_t)bc * HW2;
    float v00 = S[iy0 * W2_ + ix0], v01 = S[iy0 * W2_ + ix1];
    float v10 = S[iy1 * W2_ + ix0], v11 = S[iy1 * W2_ + ix1];
    float v = (1.0f - fy) * ((1.0f - fx) * v00 + fx * v01)
            +          fy * ((1.0f - fx) * v10 + fx * v11);
    inter[idx] = hid1[idx] + v;
}

// ---------------------------------------------------------------------------
// Stage 3b: head GEMM at LOW resolution (linear conv commuted past x8 resize):
// lsmall[b][o][p] = W3[o][:] . inter[b][:][p] + b3[o].  M padded to 16 via
// TDM out-of-bounds zero-fill (tensor_d1 = 11 rows, tile_d1 = 16 rows).
// ---------------------------------------------------------------------------
__global__ __launch_bounds__(256)
void head_gemm_wmma(const float* __restrict__ inter,  // [B][64][HW1]
                    const float* __restrict__ w3,     // [11][64]
                    const float* __restrict__ b3,     // [11]
                    float* __restrict__ lsmall) {     // [B][11][HW1]
    __shared__ float Ws[16 * HIDD];   // 4 KB, rows 11..15 zero-filled by TDM
    const int tid  = threadIdx.x;
    const int wave = tid >> 5;
    const int lane = tid & 31;
    const int half = lane >> 4;
    const int lr   = lane & 15;

    if (wave == 0) {
        tdm_load_2d(w3, (unsigned int)(size_t)(void*)Ws,
                    /*tensor_d0=*/HIDD, /*tensor_d1=*/NCL,   // 11 valid rows
                    /*stride0=*/HIDD, /*tile_d0=*/HIDD, /*tile_d1=*/16, 2u);
        __builtin_amdgcn_s_wait_tensorcnt(0);
    }
    __syncthreads();

    const int b = blockIdx.y;
    int p0 = blockIdx.x * (8 * 32) + wave * 32;
    if (p0 + 32 > HW1) p0 = HW1 - 32;

    const float* X = inter + (size_t)b * HIDD * HW1;
    v8f acc0 = {}, acc1 = {};
    const float* xrow = X + (size_t)(2 * half) * HW1 + p0 + lr;
#pragma unroll
    for (int k = 0; k < HIDD; k += 4) {
        const int kk = k + 2 * half;
        v2f a, bf0, bf1;
        a[0] = Ws[lr * HIDD + kk];
        a[1] = Ws[lr * HIDD + kk + 1];
        bf0[0] = xrow[0];   bf0[1] = xrow[HW1];
        bf1[0] = xrow[16];  bf1[1] = xrow[HW1 + 16];
        acc0 = __builtin_amdgcn_wmma_f32_16x16x4_f32(
            false, a, false, bf0, (short)0, acc0, false, false);
        acc1 = __builtin_amdgcn_wmma_f32_16x16x4_f32(
            false, a, false, bf1, (short)0, acc1, false, false);
        xrow += 4 * (size_t)HW1;
    }
    float* Yo = lsmall + (size_t)b * NCL * HW1;
#pragma unroll
    for (int v = 0; v < 8; ++v) {
        const int row = v + 8 * half;
        if (row < NCL) {
            float* yr = Yo + (size_t)row * HW1 + p0 + lr;
            yr[0]  = acc0[v] + b3[row];   // relu happens after upsample
            yr[16] = acc1[v] + b3[row];
        }
    }
}

// ---------------------------------------------------------------------------
// Stage 4 init: bbox stats [B][9] x {count, xmin, xmax, ymin, ymax}
// ---------------------------------------------------------------------------
__global__ void init_stats(int* __restrict__ stats) {
    const int i = threadIdx.x;
    const int total = B_ * (NC - 1) * 5;
    if (i >= total) return;
    const int j = i % 5;
    int v = 0;
    if (j == 1) v = OW;        // xmin
    else if (j == 2) v = -1;   // xmax
    else if (j == 3) v = OH;   // ymin
    else if (j == 4) v = -1;   // ymax
    stats[i] = v;
}

// ---------------------------------------------------------------------------
// Stage 4: x8 bilerp of 11 low-res logits, relu, softmax, argmax, bbox stats.
// grid = B*OH*OW/256 exactly; a block never straddles batches.
// ---------------------------------------------------------------------------
__global__ __launch_bounds__(256)
void upsample8x_softmax_argmax(const float* __restrict__ lsmall, // [B][11][HW1]
                               float* __restrict__ prob,         // [B][11][OHW]
                               float* __restrict__ seg,          // [B][OHW]
                               int* __restrict__ stats) {
    __shared__ int s_cnt[NC - 1], s_xmn[NC - 1], s_xmx[NC - 1],
                   s_ymn[NC - 1], s_ymx[NC - 1];
    const int tid = threadIdx.x;
    if (tid < NC - 1) {
        s_cnt[tid] = 0;  s_xmn[tid] = OW;  s_xmx[tid] = -1;
        s_ymn[tid] = OH; s_ymx[tid] = -1;
    }
    __syncthreads();

    const int idx = blockIdx.x * 256 + tid;
    const int b = idx / OHW;
    const int p = idx % OHW;
    const int x = p % OW, y = p / OW;

    float sx = (x + 0.5f) * 0.125f - 0.5f;
    float sy = (y + 0.5f) * 0.125f - 0.5f;
    int ix0 = (int)floorf(sx);  float fx = sx - (float)ix0;
    int iy0 = (int)floorf(sy);  float fy = sy - (float)iy0;
    int ix1 = ix0 + 1, iy1 = iy0 + 1;
    ix0 = min(max(ix0, 0), W1_ - 1); ix1 = min(max(ix1, 0), W1_ - 1);
    iy0 = min(max(iy0, 0), H1 - 1);  iy1 = min(max(iy1, 0), H1 - 1);

    const float w00 = (1.0f - fy) * (1.0f - fx);
    const float w01 = (1.0f - fy) * fx;
    const float w10 = fy * (1.0f - fx);
    const float w11 = fy * fx;
    const int i00 = iy0 * W1_ + ix0, i01 = iy0 * W1_ + ix1;
    const int i10 = iy1 * W1_ + ix0, i11 = iy1 * W1_ + ix1;

    const float* L = lsmall + (size_t)b * NCL * HW1;
    float vals[NCL];
    float maxv = -1.0f;   // relu output >= 0
    int amax = 0;
#pragma unroll
    for (int c = 0; c < NCL; ++c) {
        const float* Lc = L + (size_t)c * HW1;
        float v = w00 * Lc[i00] + w01 * Lc[i01] + w10 * Lc[i10] + w11 * Lc[i11];
        v = fmaxf(v, 0.0f);                    // relu AFTER upsample (== reference)
        vals[c] = v;
        if (v > maxv) { maxv = v; amax = c; }  // strict > => first max (jnp.argmax)
    }
    float sum = 0.0f;
#pragma unroll
    for (int c = 0; c < NCL; ++c) {
        float e = __expf(vals[c] - maxv);
        vals[c] = e;
        sum += e;
    }
    const float inv = 1.0f / sum;
    float* Pb = prob + (size_t)b * NCL * OHW + p;
#pragma unroll
    for (int c = 0; c < NCL; ++c) {
        Pb[(size_t)c * OHW] = vals[c] * inv;
    }
    seg[idx] = (float)amax;

    if (amax >= 1) {
        const int ci = amax - 1;
        atomicAdd(&s_cnt[ci], 1);
        atomicMin(&s_xmn[ci], x); atomicMax(&s_xmx[ci], x);
        atomicMin(&s_ymn[ci], y); atomicMax(&s_ymx[ci], y);
    }
    __syncthreads();

    if (tid < NC - 1) {
        const int ci = tid;
        if (s_cnt[ci] > 0) {
            int* st = stats + ((size_t)b * (NC - 1) + ci) * 5;
            atomicAdd(&st[0], s_cnt[ci]);
            atomicMin(&st[1], s_xmn[ci]); atomicMax(&st[2], s_xmx[ci]);
            atomicMin(&st[3], s_ymn[ci]); atomicMax(&st[4], s_ymx[ci]);
        }
    }
}

// ---------------------------------------------------------------------------
// Stage 5: finalize bbx (36 x 6): [bid, xmin, ymin, xmax, ymax, cls] or -1
// ---------------------------------------------------------------------------
__global__ void bbox_finalize(const int* __restrict__ stats,
                              float* __restrict__ bbx) {
    const int r = threadIdx.x;
    if (r >= B_ * (NC - 1)) return;
    const int b = r / (NC - 1);
    const int ci = r % (NC - 1);
    const int* st = stats + (size_t)r * 5;
    float* o = bbx + (size_t)r * 6;
    if (st[0] >= 100) {          // THRESH
        o[0] = (float)b;
        o[1] = (float)st[1];     // xmin
        o[2] = (float)st[3];     // ymin
        o[3] = (float)st[2];     // xmax
        o[4] = (float)st[4];     // ymax
        o[5] = (float)(ci + 1);  // class
    } else {
        o[0] = o[1] = o[2] = o[3] = o[4] = o[5] = -1.0f;
    }
}

// ---------------------------------------------------------------------------
extern "C" void kernel_launch(void* const* d_in, const int* in_sizes, int n_in,
                              void* d_out, int out_size, void* d_ws, size_t ws_size,
                              hipStream_t stream) {
    const float* feature1 = (const float*)d_in[0];  // [4][512][60][80]
    const float* feature2 = (const float*)d_in[1];  // [4][512][30][40]
    const float* w1 = (const float*)d_in[2];        // [64][512]
    const float* b1 = (const float*)d_in[3];        // [64]
    const float* w2 = (const float*)d_in[4];        // [64][512]
    const float* b2 = (const float*)d_in[5];        // [64]
    const float* w3 = (const float*)d_in[6];        // [11][64]
    const float* b3 = (const float*)d_in[7];        // [11]

    float* prob = (float*)d_out;                          // 4*11*480*640
    float* seg  = prob + (size_t)B_ * NCL * OHW;
    float* bbx  = seg + (size_t)B_ * OHW;

    float* ws     = (float*)d_ws;
    float* hid1   = ws;                                      // 4*64*4800
    float* hid2   = hid1 + (size_t)B_ * HIDD * HW1;          // 4*64*1200
    float* inter  = hid2 + (size_t)B_ * HIDD * HW2;          // 4*64*4800
    float* lsmall = inter + (size_t)B_ * HIDD * HW1;         // 4*11*4800
    int*   stats  = (int*)(lsmall + (size_t)B_ * NCL * HW1); // 4*9*5 ints

    const dim3 blk(256);

    // conv1: 4800 px / (8 waves * 32 px) = 18.75 -> 19 blocks (clamped tail)
    conv1x1_relu_wmma<<<dim3(19, B_), blk, 0, stream>>>(feature1, w1, b1, hid1, HW1);
    // conv2: 1200 px -> 5 blocks
    conv1x1_relu_wmma<<<dim3(5, B_), blk, 0, stream>>>(feature2, w2, b2, hid2, HW2);

    upsample2x_add<<<(B_ * HIDD * HW1 + 255) / 256, blk, 0, stream>>>(hid1, hid2, inter);

    head_gemm_wmma<<<dim3(19, B_), blk, 0, stream>>>(inter, w3, b3, lsmall);

    init_stats<<<1, 256, 0, stream>>>(stats);

    upsample8x_softmax_argmax<<<(B_ * OHW) / 256, blk, 0, stream>>>(lsmall, prob, seg, stats);

    bbox_finalize<<<1, 64, 0, stream>>>(stats, bbx);
}